// MuyGPs_layer_27015344292373
// MI455X (gfx1250) — compile-verified
//
#include <hip/hip_runtime.h>
#include <math.h>

// MuyGPs layer for MI455X (gfx1250): one workgroup per batch element.
// Gram matrix via V_WMMA_F32_16X16X4_F32, Matern-1.5 transform, in-LDS
// Cholesky solve (64x64, 17 RHS), deterministic cross-batch reduction.

#define EPSV   1e-5f
#define SQRT3F 1.7320508075688772f
#define DIM    128
#define BATCH  8192
#define KNN    64
#define RR     16
#define NC     17           // RHS columns = 16 targets + Kcross

// LDS strides (bank-conflict padding; 64 banks x 4B)
#define SP 132              // P row stride: (132*m + k) % 64 = (4m+k)%64 -> conflict-free
#define SG 65               // G row stride: (65*i + j) % 64 = (i+j)%64   -> conflict-free cols

typedef __attribute__((ext_vector_type(2))) float v2f;
typedef __attribute__((ext_vector_type(8))) float v8f;

__global__ __launch_bounds__(256)
void muygps_batch_kernel(const float* __restrict__ x,
                         const int*   __restrict__ batch_indices,
                         const int*   __restrict__ nn_indices,
                         const float* __restrict__ nn_targets,
                         const float* __restrict__ length_scale,
                         float* __restrict__ predictions,   // [B][16]
                         float* __restrict__ diag_var,      // [B]
                         float* __restrict__ sigma_partial) // [B][16]
{
    __shared__ float P[KNN * SP];     // gathered neighbor points, 64x128 (padded)
    __shared__ float G[KNN * SG];     // gram -> Kmat -> Cholesky factor (padded)
    __shared__ float RHS[KNN * NC];   // 64x17 rhs -> solution
    __shared__ float TGT[KNN * RR];   // copy of targets (RHS is overwritten)
    __shared__ float qv[DIM];
    __shared__ float sq[KNN];
    __shared__ float Kc[KNN];
    __shared__ float dinv[KNN];

    const int b    = blockIdx.x;
    const int tid  = threadIdx.x;
    const int lane = tid & 31;
    const int wave = tid >> 5;

    // ---- gather neighbor rows + query point + targets into LDS ----
    const int* nnrow = nn_indices + (size_t)b * KNN;
    for (int e = tid; e < KNN * (DIM / 4); e += 256) {
        const int row = e >> 5;           // DIM/4 = 32 float4 per row
        const int c4  = e & 31;
        const float4 v = ((const float4*)(x + (size_t)nnrow[row] * DIM))[c4];
        float* dst = &P[row * SP + c4 * 4];
        dst[0] = v.x; dst[1] = v.y; dst[2] = v.z; dst[3] = v.w;
    }
    if (tid < DIM) qv[tid] = x[(size_t)batch_indices[b] * DIM + tid];
    for (int e = tid; e < KNN * RR; e += 256)
        TGT[e] = nn_targets[(size_t)b * KNN * RR + e];
    __syncthreads();

    // ---- G = P * P^T via v_wmma_f32_16x16x4_f32 ----
    // Wave w owns tiles (ti0, tj) and (ti0+2, tj); both share the B fragment.
    const int tj   = wave & 3;
    const int ti0  = wave >> 2;
    const int ti1  = ti0 + 2;
    const int m    = lane & 15;
    const int half = lane >> 4;           // 0: K pair {0,1}, 1: K pair {2,3}
    const int koff = half * 2;

    v8f c0 = {0.f,0.f,0.f,0.f,0.f,0.f,0.f,0.f};
    v8f c1 = c0;
    const float* pA0 = &P[(ti0 * 16 + m) * SP + koff];
    const float* pA1 = &P[(ti1 * 16 + m) * SP + koff];
    const float* pB  = &P[(tj  * 16 + m) * SP + koff];

    for (int k0 = 0; k0 < DIM; k0 += 4) {
        v2f bf = { pB[k0],  pB[k0 + 1]  };   // B = P^T tile (same LDS pattern as A)
        v2f a0 = { pA0[k0], pA0[k0 + 1] };
        v2f a1 = { pA1[k0], pA1[k0 + 1] };
        c0 = __builtin_amdgcn_wmma_f32_16x16x4_f32(false, a0, false, bf,
                                                   (short)0, c0, false, false);
        c1 = __builtin_amdgcn_wmma_f32_16x16x4_f32(false, a1, false, bf,
                                                   (short)0, c1, false, false);
    }
    // C/D layout: VGPR v -> M = v (lanes 0-15) or v+8 (lanes 16-31), N = lane%16
    {
        const int colg  = tj * 16 + m;
        const int rbase = half * 8;
        #pragma unroll
        for (int v = 0; v < 8; ++v) {
            G[(ti0 * 16 + rbase + v) * SG + colg] = c0[v];
            G[(ti1 * 16 + rbase + v) * SG + colg] = c1[v];
        }
    }
    __syncthreads();

    const float inv_ls = 1.0f / length_scale[0];

    // ---- squared norms (= gram diagonal), cross distances, Kcross ----
    if (tid < KNN) {
        sq[tid] = G[tid * SG + tid];
        float acc = 0.f;
        const float* pr = &P[tid * SP];
        for (int d = 0; d < DIM; ++d) {
            const float df = qv[d] - pr[d];
            acc += df * df;
        }
        const float dist = sqrtf(fmaxf(acc, 0.f));
        const float s = SQRT3F * dist * inv_ls;
        Kc[tid] = (1.f + s) * expf(-s);
    }
    __syncthreads();

    // ---- Matern-1.5 transform of gram in place (+ EPS on diagonal) ----
    for (int e = tid; e < KNN * KNN; e += 256) {
        const int i = e >> 6, j = e & 63;
        const float d2   = sq[i] + sq[j] - 2.f * G[i * SG + j];
        const float dist = sqrtf(fmaxf(d2, 0.f));
        const float s    = SQRT3F * dist * inv_ls;
        float v = (1.f + s) * expf(-s);
        if (i == j) v += EPSV;
        G[i * SG + j] = v;
    }
    // ---- RHS = [targets | Kcross] ----
    for (int e = tid; e < KNN * NC; e += 256) {
        const int i = e / NC, c = e % NC;
        RHS[e] = (c < RR) ? TGT[i * RR + c] : Kc[i];
    }
    __syncthreads();

    // ---- Cholesky (in place, full symmetric update) ----
    for (int k = 0; k < KNN; ++k) {
        if (tid == 0) {
            const float v = sqrtf(G[k * SG + k]);
            G[k * SG + k] = v;
            dinv[k] = 1.0f / v;
        }
        __syncthreads();
        for (int i = k + 1 + tid; i < KNN; i += 256) G[i * SG + k] *= dinv[k];
        __syncthreads();
        const int n = KNN - 1 - k;
        for (int e = tid; e < n * n; e += 256) {
            const int i = k + 1 + e / n, j = k + 1 + e % n;
            G[i * SG + j] -= G[i * SG + k] * G[j * SG + k];
        }
        __syncthreads();
    }

    // ---- forward solve: L y = RHS ----
    for (int j = 0; j < KNN; ++j) {
        if (tid < NC) RHS[j * NC + tid] *= dinv[j];
        __syncthreads();
        const int rows = KNN - 1 - j;
        for (int e = tid; e < rows * NC; e += 256) {
            const int i = j + 1 + e / NC, c = e % NC;
            RHS[i * NC + c] -= G[i * SG + j] * RHS[j * NC + c];
        }
        __syncthreads();
    }
    // ---- back solve: L^T x = y ----
    for (int j = KNN - 1; j >= 0; --j) {
        if (tid < NC) RHS[j * NC + tid] *= dinv[j];
        __syncthreads();
        for (int e = tid; e < j * NC; e += 256) {
            const int i = e / NC, c = e % NC;           // i < j
            RHS[i * NC + c] -= G[j * SG + i] * RHS[j * NC + c];
        }
        __syncthreads();
    }

    // ---- per-batch outputs ----
    if (tid < NC) {
        float acc = 0.f;
        for (int k = 0; k < KNN; ++k) acc += Kc[k] * RHS[k * NC + tid];
        if (tid < RR) predictions[(size_t)b * RR + tid] = acc;
        else          diag_var[b] = 1.0f - acc;
    }
    if (tid >= 32 && tid < 48) {      // separate wave: sigma_sq partials
        const int r = tid - 32;
        float acc = 0.f;
        for (int k = 0; k < KNN; ++k) acc += TGT[k * RR + r] * RHS[k * NC + r];
        sigma_partial[(size_t)b * RR + r] = acc;
    }
}

// Deterministic fixed-order reduction of sigma partials (single block).
__global__ __launch_bounds__(256)
void sigma_reduce_kernel(const float* __restrict__ partial,
                         float* __restrict__ sigma_out,
                         float* __restrict__ sigma_ws)
{
    __shared__ float red[16][17];
    const int tid = threadIdx.x;
    const int r = tid & 15, chunk = tid >> 4;      // 16 chunks x 16 cols
    float acc = 0.f;
    for (int b = chunk; b < BATCH; b += 16) acc += partial[b * RR + r];
    red[chunk][r] = acc;
    __syncthreads();
    if (tid < RR) {
        float s = 0.f;
        for (int c = 0; c < 16; ++c) s += red[c][tid];
        s /= (float)KNN * (float)BATCH;
        sigma_out[tid] = s;
        sigma_ws[tid]  = s;
    }
}

__global__ __launch_bounds__(256)
void variance_kernel(const float* __restrict__ diag_var,
                     const float* __restrict__ sigma,
                     float* __restrict__ variances)
{
    const int idx = blockIdx.x * 256 + threadIdx.x;
    if (idx < BATCH * RR) {
        const int b = idx >> 4, r = idx & 15;
        variances[idx] = diag_var[b] * sigma[r];
    }
}

extern "C" void kernel_launch(void* const* d_in, const int* in_sizes, int n_in,
                              void* d_out, int out_size, void* d_ws, size_t ws_size,
                              hipStream_t stream)
{
    const float* x     = (const float*)d_in[0];   // [N, D]
    const int*   bidx  = (const int*)  d_in[1];   // [B]
    const int*   nnidx = (const int*)  d_in[2];   // [B, K]
    const float* tgt   = (const float*)d_in[3];   // [B, K, R]
    const float* ls    = (const float*)d_in[4];   // scalar

    float* out         = (float*)d_out;
    float* predictions = out;                     // B*R
    float* variances   = out + BATCH * RR;        // B*R
    float* sigma_out   = out + 2 * BATCH * RR;    // R

    float* ws        = (float*)d_ws;
    float* diag_var  = ws;                        // B floats
    float* sigma_ws  = ws + BATCH;                // 16 floats
    float* partial   = ws + BATCH + 256;          // B*R floats (aligned)

    muygps_batch_kernel<<<BATCH, 256, 0, stream>>>(x, bidx, nnidx, tgt, ls,
                                                   predictions, diag_var, partial);
    sigma_reduce_kernel<<<1, 256, 0, stream>>>(partial, sigma_out, sigma_ws);
    variance_kernel<<<(BATCH * RR + 255) / 256, 256, 0, stream>>>(diag_var, sigma_ws, variances);
}